// LightGCN_49675591745779
// MI455X (gfx1250) — compile-verified
//
#include <hip/hip_runtime.h>
#include <hip/hip_bf16.h>

#define EMBED 64

// ---------------- COO SpMM: y[row] += val * x[col], 64-dim rows ----------------
//
// 256 threads = 8 wave32. Each wave owns 32 edges of the block's 256-edge tile.
// Edge triples are staged into LDS with CDNA5 async loads (ASYNCcnt-tracked
// global_load_async_to_lds_b32), completed wave-locally with s_wait_asynccnt 0
// (no workgroup barrier needed: each wave reads only its own 32-entry slice).
// Then each half-wave of 16 lanes processes one edge: lane sub (0..15) handles
// float4 chunk sub of the 64-float embedding row (coalesced 256B gather), and
// scatters with native no-return global_atomic_add_f32 (STOREcnt-tracked).

__global__ __launch_bounds__(256) void spmm_coo(
    const int* __restrict__ rows, const int* __restrict__ cols,
    const float* __restrict__ vals, const float* __restrict__ x,
    float* __restrict__ y, int num_edges) {
  __shared__ int   s_row[256];
  __shared__ int   s_col[256];
  __shared__ float s_val[256];

  const int tid  = threadIdx.x;
  const int wave = tid >> 5;     // 0..7
  const int lane = tid & 31;     // 0..31
  const int grp  = lane >> 4;    // which of 2 concurrent edges in this wave
  const int sub  = lane & 15;    // float4 chunk within 64-float row

  const int ebase = blockIdx.x * 256 + wave * 32;  // this wave's 32 edges
  const int eld   = ebase + lane;

  if (eld < num_edges) {
    // addr[31:0] of a generic LDS pointer == wave-relative LDS byte address
    // (ISA 10.2: flat->LDS aperture uses addr[31:0] directly).
    unsigned lr = (unsigned)(unsigned long long)&s_row[tid];
    unsigned lc = (unsigned)(unsigned long long)&s_col[tid];
    unsigned lv = (unsigned)(unsigned long long)&s_val[tid];
    const int*   gr = rows + eld;
    const int*   gc = cols + eld;
    const float* gv = vals + eld;
    asm volatile("global_load_async_to_lds_b32 %0, %1, off" :: "v"(lr), "v"(gr) : "memory");
    asm volatile("global_load_async_to_lds_b32 %0, %1, off" :: "v"(lc), "v"(gc) : "memory");
    asm volatile("global_load_async_to_lds_b32 %0, %1, off" :: "v"(lv), "v"(gv) : "memory");
  }
  // Wave-local staging: only this wave reads what it wrote -> no barrier needed.
  asm volatile("s_wait_asynccnt 0x0" ::: "memory");

  const int sb = wave * 32;
#pragma unroll 4
  for (int it = 0; it < 16; ++it) {
    const int el = it * 2 + grp;       // 0..31 within wave tile
    const int e  = ebase + el;
    if (e < num_edges) {
      const int   r = s_row[sb + el];   // ds_load_b32 (broadcast across 16 lanes)
      const int   c = s_col[sb + el];
      const float v = s_val[sb + el];
      const float4 xv = *(const float4*)(x + (size_t)c * EMBED + sub * 4);
      float4 m;
      m.x = xv.x * v; m.y = xv.y * v; m.z = xv.z * v; m.w = xv.w * v;
      float* yp = y + (size_t)r * EMBED + sub * 4;
      asm volatile(
          "global_atomic_add_f32 %0, %1, off\n\t"
          "global_atomic_add_f32 %0, %2, off offset:4\n\t"
          "global_atomic_add_f32 %0, %3, off offset:8\n\t"
          "global_atomic_add_f32 %0, %4, off offset:12"
          :: "v"(yp), "v"(m.x), "v"(m.y), "v"(m.z), "v"(m.w)
          : "memory");
    }
  }
}

// ---------------- elementwise kernels (float4 vectorized) ----------------

__global__ __launch_bounds__(256) void init_concat(
    const float* __restrict__ ue, const float* __restrict__ ie,
    float* __restrict__ cur, float* __restrict__ acc,
    int user_elems, int total_elems) {
  int i = (blockIdx.x * blockDim.x + threadIdx.x) * 4;
  if (i >= total_elems) return;
  float4 v = (i < user_elems) ? *(const float4*)(ue + i)
                              : *(const float4*)(ie + (i - user_elems));
  *(float4*)(cur + i) = v;
  *(float4*)(acc + i) = v;
}

__global__ __launch_bounds__(256) void zero_buf(float* __restrict__ p, int total_elems) {
  int i = (blockIdx.x * blockDim.x + threadIdx.x) * 4;
  if (i < total_elems) {
    float4 z; z.x = 0.f; z.y = 0.f; z.z = 0.f; z.w = 0.f;
    *(float4*)(p + i) = z;
  }
}

// acc += nxt, and zero `zbuf` (the buffer that becomes next layer's atomic
// target) in the same pass — saves a full 38.4MB sweep per layer.
__global__ __launch_bounds__(256) void axpy_zero(
    float* __restrict__ acc, const float* __restrict__ nxt,
    float* __restrict__ zbuf, int total_elems) {
  int i = (blockIdx.x * blockDim.x + threadIdx.x) * 4;
  if (i < total_elems) {
    float4 a = *(float4*)(acc + i);
    float4 b = *(const float4*)(nxt + i);
    a.x += b.x; a.y += b.y; a.z += b.z; a.w += b.w;
    *(float4*)(acc + i) = a;
    float4 z; z.x = 0.f; z.y = 0.f; z.z = 0.f; z.w = 0.f;
    *(float4*)(zbuf + i) = z;
  }
}

// Final layer: acc = (acc + nxt) * s
__global__ __launch_bounds__(256) void axpy_scale(
    float* __restrict__ acc, const float* __restrict__ nxt,
    float s, int total_elems) {
  int i = (blockIdx.x * blockDim.x + threadIdx.x) * 4;
  if (i < total_elems) {
    float4 a = *(float4*)(acc + i);
    float4 b = *(const float4*)(nxt + i);
    a.x = (a.x + b.x) * s; a.y = (a.y + b.y) * s;
    a.z = (a.z + b.z) * s; a.w = (a.w + b.w) * s;
    *(float4*)(acc + i) = a;
  }
}

// ---------------- launch ----------------

extern "C" void kernel_launch(void* const* d_in, const int* in_sizes, int n_in,
                              void* d_out, int out_size, void* d_ws, size_t ws_size,
                              hipStream_t stream) {
  const float* ue = (const float*)d_in[0];   // user_emb  (NUM_USERS*64)
  const float* ie = (const float*)d_in[1];   // item_emb  (NUM_ITEMS*64)
  const float* ev = (const float*)d_in[2];   // edge_vals (NUM_EDGES)
  const int*   er = (const int*)d_in[3];     // edge_row
  const int*   ec = (const int*)d_in[4];     // edge_col

  const int user_elems = in_sizes[0];
  const int item_elems = in_sizes[1];
  const int num_edges  = in_sizes[2];
  const int total      = user_elems + item_elems;   // == out_size (N_NODES*64)

  float* acc = (float*)d_out;       // accumulator lives in the output buffer
  float* cur = (float*)d_ws;        // 38.4 MB
  float* nxt = cur + total;         // 38.4 MB

  const int T = 256;
  const int eblocks = (total / 4 + T - 1) / T;          // elementwise grids
  const int sblocks = (num_edges + 255) / 256;          // 256 edges per block

  // cur = concat(ue, ie); acc = cur
  init_concat<<<eblocks, T, 0, stream>>>(ue, ie, cur, acc, user_elems, total);
  // nxt must start zeroed for the first layer's atomics
  zero_buf<<<eblocks, T, 0, stream>>>(nxt, total);

  // Layer 0: nxt = A*cur ; acc += nxt ; zero(cur) ; swap
  spmm_coo<<<sblocks, 256, 0, stream>>>(er, ec, ev, cur, nxt, num_edges);
  axpy_zero<<<eblocks, T, 0, stream>>>(acc, nxt, cur, total);
  { float* t = cur; cur = nxt; nxt = t; }

  // Layer 1
  spmm_coo<<<sblocks, 256, 0, stream>>>(er, ec, ev, cur, nxt, num_edges);
  axpy_zero<<<eblocks, T, 0, stream>>>(acc, nxt, cur, total);
  { float* t = cur; cur = nxt; nxt = t; }

  // Layer 2 (final): acc = (acc + A*cur) / 4
  spmm_coo<<<sblocks, 256, 0, stream>>>(er, ec, ev, cur, nxt, num_edges);
  axpy_scale<<<eblocks, T, 0, stream>>>(acc, nxt, 0.25f, total);
}